// get_model_51144470561297
// MI455X (gfx1250) — compile-verified
//
#include <hip/hip_runtime.h>
#include <hip/hip_bf16.h>
#include <math.h>

typedef __attribute__((ext_vector_type(2))) float v2f;
typedef __attribute__((ext_vector_type(8))) float v8f;

#define BN_EPS 1e-5f
#define CDIV(a, b) (((a) + (b) - 1) / (b))
#define WMMA4(A, Bf, Cacc) __builtin_amdgcn_wmma_f32_16x16x4_f32(false, (A), false, (Bf), (short)0, (Cacc), false, false)

// ---------------- workspace layout (units: floats) ----------------
constexpr size_t OFF_PTS   = 0;                       // 16*4096*3   = 196608
constexpr size_t OFF_DIST  = OFF_PTS   + 196608;      // 16*4096     = 65536
constexpr size_t OFF_FIDX  = OFF_DIST  + 65536;       // 16*1024 int = 16384
constexpr size_t OFF_L1X   = OFF_FIDX  + 16384;       // 16*1024*3   = 49152
constexpr size_t OFF_GIDX1 = OFF_L1X   + 49152;       // 16*1024*32  = 524288 (int)
constexpr size_t OFF_L1F   = OFF_GIDX1 + 524288;      // 16*1024*64  = 1048576
constexpr size_t OFF_L2X   = OFF_L1F   + 1048576;     // 16*256*3    = 12288
constexpr size_t OFF_GIDX2 = OFF_L2X   + 12288;       // 16*256*32   = 131072 (int)
constexpr size_t OFF_L2F   = OFF_GIDX2 + 131072;      // 16*256*128  = 524288
constexpr size_t OFF_L3F   = OFF_L2F   + 524288;      // 16*512      = 8192
constexpr size_t OFF_N2    = OFF_L3F   + 8192;        // 16*256*256  = 1048576
constexpr size_t OFF_KI2   = OFF_N2    + 1048576;     // 16*1024*3   = 49152 (int)
constexpr size_t OFF_KW2   = OFF_KI2   + 49152;       // 49152
constexpr size_t OFF_N1    = OFF_KW2   + 49152;       // 16*1024*256 = 4194304
constexpr size_t OFF_KI3   = OFF_N1    + 4194304;     // 16*4096*3   = 196608 (int)
constexpr size_t OFF_KW3   = OFF_KI3   + 196608;      // 196608
constexpr size_t OFF_STATS = OFF_KW3   + 196608;      // 1024 (mean[512], var[512])
constexpr size_t OFF_WPAD  = OFF_STATS + 1024;        // padded weights/biases scratch
constexpr size_t OFF_BUF0  = OFF_WPAD  + 1200000;     // 33554432
constexpr size_t OFF_BUF1  = OFF_BUF0  + 33554432;    // 33554432

// ---------------- (B,3,N) -> (B,N,3) ----------------
__global__ void k_transpose_xyz(const float* __restrict__ xyz, float* __restrict__ pts,
                                int B, int N) {
  int i = blockIdx.x * blockDim.x + threadIdx.x;
  if (i >= B * N * 3) return;
  int c = i % 3, n = (i / 3) % N, b = i / (3 * N);
  pts[i] = xyz[((size_t)b * 3 + c) * N + n];
}

// ---------------- farthest point sampling: one workgroup per batch ----------------
__global__ __launch_bounds__(256) void k_fps(const float* __restrict__ pts,
                                             float* __restrict__ dist,
                                             int* __restrict__ fidx, int N, int np) {
  int b = blockIdx.x;
  __shared__ float sx[4096 * 3];         // 48 KB: points resident in LDS
  __shared__ float rv[256];
  __shared__ int   ri[256];
  __shared__ int   sfar;
  const float* P = pts + (size_t)b * N * 3;
  float* D = dist + (size_t)b * N;
  for (int i = threadIdx.x; i < N * 3; i += 256) sx[i] = P[i];
  for (int n = threadIdx.x; n < N; n += 256) D[n] = 1e10f;
  if (threadIdx.x == 0) sfar = 0;
  __syncthreads();
  for (int it = 0; it < np; ++it) {
    int far = sfar;
    if (threadIdx.x == 0) fidx[b * np + it] = far;
    float cx = sx[far * 3], cy = sx[far * 3 + 1], cz = sx[far * 3 + 2];
    float bv = -1.0f; int bi = 0;
    for (int n = threadIdx.x; n < N; n += 256) {
      float dx = sx[n * 3] - cx, dy = sx[n * 3 + 1] - cy, dz = sx[n * 3 + 2] - cz;
      float d = fminf(D[n], dx * dx + dy * dy + dz * dz);
      D[n] = d;
      if (d > bv) { bv = d; bi = n; }
    }
    rv[threadIdx.x] = bv; ri[threadIdx.x] = bi;
    __syncthreads();
    for (int off = 128; off > 0; off >>= 1) {
      if ((int)threadIdx.x < off) {
        float v2v = rv[threadIdx.x + off]; int j = ri[threadIdx.x + off];
        if (v2v > rv[threadIdx.x] || (v2v == rv[threadIdx.x] && j < ri[threadIdx.x])) {
          rv[threadIdx.x] = v2v; ri[threadIdx.x] = j;
        }
      }
      __syncthreads();
    }
    if (threadIdx.x == 0) sfar = ri[0];
    __syncthreads();
  }
}

// ---------------- gather sampled coordinates ----------------
__global__ void k_gather_xyz(const float* __restrict__ pts, const int* __restrict__ fidx,
                             float* __restrict__ out, int B, int N, int S) {
  int i = blockIdx.x * blockDim.x + threadIdx.x;
  if (i >= B * S * 3) return;
  int c = i % 3, s = (i / 3) % S, b = i / (3 * S);
  out[i] = pts[((size_t)b * N + fidx[b * S + s]) * 3 + c];
}

// ---------------- ball query: one workgroup per batch, LDS-resident cloud ----------------
__global__ __launch_bounds__(256) void k_ball_query(const float* __restrict__ xyz,
                                                    const float* __restrict__ nxyz,
                                                    int* __restrict__ gidx,
                                                    int N, int S, float r2, int K) {
  int b = blockIdx.x;
  __shared__ float sx[4096 * 3];
  const float* P = xyz + (size_t)b * N * 3;
  for (int i = threadIdx.x; i < N * 3; i += 256) sx[i] = P[i];
  __syncthreads();
  for (int s = threadIdx.x; s < S; s += 256) {
    const float* q = nxyz + ((size_t)b * S + s) * 3;
    float qx = q[0], qy = q[1], qz = q[2];
    int* out = gidx + ((size_t)b * S + s) * K;
    int cnt = 0, firstIdx = 0, bn = 0;
    float bd = 3.4e38f;
    for (int n = 0; n < N; ++n) {
      float dx = sx[n * 3] - qx, dy = sx[n * 3 + 1] - qy, dz = sx[n * 3 + 2] - qz;
      float d = dx * dx + dy * dy + dz * dz;
      if (d < bd) { bd = d; bn = n; }
      if (d <= r2 && cnt < K) { if (cnt == 0) firstIdx = n; out[cnt++] = n; }
    }
    int fill = (cnt > 0) ? firstIdx : bn;
    for (int k = cnt; k < K; ++k) out[k] = fill;
  }
}

// ---------------- gather grouped features, write with padded channel stride ----------------
__global__ void k_gather_group(const float* __restrict__ feat, const int* __restrict__ gidx,
                               float* __restrict__ out, int B, int N, int S, int K,
                               int C, int Cp) {
  int i = blockIdx.x * blockDim.x + threadIdx.x;
  if (i >= B * S * K * Cp) return;
  int c = i % Cp; int t = i / Cp;
  int k = t % K; t /= K;
  int s = t % S; int b = t / S;
  out[i] = (c < C)
     ? feat[((size_t)b * N + gidx[((size_t)b * S + s) * K + k]) * C + c]
     : 0.f;
}

// ---------------- pad weights [Cout,Cin] -> [Coutp,Cinp] and bias -> [Coutp] ----------------
__global__ void k_pad_wb(const float* __restrict__ w, const float* __restrict__ b,
                         float* __restrict__ wp, float* __restrict__ bp,
                         int Cout, int Cin, int Coutp, int Cinp) {
  int i = blockIdx.x * blockDim.x + threadIdx.x;
  int tot = Coutp * Cinp;
  if (i < tot) {
    int r = i / Cinp, c = i % Cinp;
    wp[i] = (r < Cout && c < Cin) ? w[(size_t)r * Cin + c] : 0.f;
  }
  if (i < Coutp) bp[i] = (i < Cout) ? b[i] : 0.f;
}

// ---------------- WMMA f32 GEMM, guard-free: Y[M,Coutp] = X[M,Cinp] * Wp^T + bias ----------
// Requirements (guaranteed by padding): M % 64 == 0, Cinp % 16 == 0, Coutp % 16 == 0.
// Block = 8 waves sharing one N-tile whose 16xCinp weight tile is staged in LDS once.
// Each wave computes a 64x16 supertile; K is processed in chunks of 16 (4 WMMA steps):
// all 16 A-fragment loads + B LDS loads for a chunk are issued up front, then 16
// back-to-back WMMAs consume them, amortizing the load-wait 16x.
__global__ __launch_bounds__(256) void k_gemm_bias(const float* __restrict__ X,
                                                   const float* __restrict__ Wp,
                                                   const float* __restrict__ bias,
                                                   float* __restrict__ Y,
                                                   int M, int Cinp, int Coutp) {
  extern __shared__ float sw[];                       // 16 * Cinp floats
  int n0 = (int)blockIdx.x << 4;
  int wtile = Cinp << 4;
  for (int i = threadIdx.x; i < wtile; i += 256)
    sw[i] = Wp[(size_t)(n0 + i / Cinp) * Cinp + (i % Cinp)];
  __syncthreads();

  int tm = (int)blockIdx.y * 8 + ((int)threadIdx.x >> 5);
  if (tm >= (M >> 6)) return;                          // wave-uniform: EXEC stays all-ones
  int lane  = threadIdx.x & 31;
  int r16   = lane & 15;
  int khalf = lane >> 4;
  int m0 = tm << 6;
  // A fragment: lane holds row lane%16, K pair 2*(lane/16); B mirrors with N=lane%16.
  const v2f* wl = (const v2f*)(sw + (size_t)r16 * Cinp) + khalf;
  const v2f* x0 = (const v2f*)(X + (size_t)(m0 +  0 + r16) * Cinp) + khalf;
  const v2f* x1 = (const v2f*)(X + (size_t)(m0 + 16 + r16) * Cinp) + khalf;
  const v2f* x2 = (const v2f*)(X + (size_t)(m0 + 32 + r16) * Cinp) + khalf;
  const v2f* x3 = (const v2f*)(X + (size_t)(m0 + 48 + r16) * Cinp) + khalf;
  v8f acc0 = {}, acc1 = {}, acc2 = {}, acc3 = {};
  int chunks = Cinp >> 4;                              // 4 K-steps (K=16 floats) per chunk
  for (int kc = 0; kc < chunks; ++kc) {
    int o = kc << 3;                                   // v2f index of this chunk's step 0
    // ---- batched loads: 4 B fragments (LDS) + 16 A fragments (global)
    v2f b0 = wl[o], b1 = wl[o + 2], b2 = wl[o + 4], b3 = wl[o + 6];
    v2f a00 = x0[o], a01 = x0[o + 2], a02 = x0[o + 4], a03 = x0[o + 6];
    v2f a10 = x1[o], a11 = x1[o + 2], a12 = x1[o + 4], a13 = x1[o + 6];
    v2f a20 = x2[o], a21 = x2[o + 2], a22 = x2[o + 4], a23 = x2[o + 6];
    v2f a30 = x3[o], a31 = x3[o + 2], a32 = x3[o + 4], a33 = x3[o + 6];
    // ---- 16 back-to-back WMMAs
    acc0 = WMMA4(a00, b0, acc0); acc1 = WMMA4(a10, b0, acc1);
    acc2 = WMMA4(a20, b0, acc2); acc3 = WMMA4(a30, b0, acc3);
    acc0 = WMMA4(a01, b1, acc0); acc1 = WMMA4(a11, b1, acc1);
    acc2 = WMMA4(a21, b1, acc2); acc3 = WMMA4(a31, b1, acc3);
    acc0 = WMMA4(a02, b2, acc0); acc1 = WMMA4(a12, b2, acc1);
    acc2 = WMMA4(a22, b2, acc2); acc3 = WMMA4(a32, b2, acc3);
    acc0 = WMMA4(a03, b3, acc0); acc1 = WMMA4(a13, b3, acc1);
    acc2 = WMMA4(a23, b3, acc2); acc3 = WMMA4(a33, b3, acc3);
  }

  float bv = bias[n0 + r16];
  // C/D layout: VGPR r -> row r + 8*khalf, col = lane%16
  float* yb = Y + (size_t)(m0 + (khalf << 3)) * Coutp + (n0 + r16);
#pragma unroll
  for (int r = 0; r < 8; ++r) yb[(size_t)(r +  0) * Coutp] = acc0[r] + bv;
#pragma unroll
  for (int r = 0; r < 8; ++r) yb[(size_t)(r + 16) * Coutp] = acc1[r] + bv;
#pragma unroll
  for (int r = 0; r < 8; ++r) yb[(size_t)(r + 32) * Coutp] = acc2[r] + bv;
#pragma unroll
  for (int r = 0; r < 8; ++r) yb[(size_t)(r + 48) * Coutp] = acc3[r] + bv;
}

// ---------------- batchnorm statistics (deterministic, block per channel) ----------------
__global__ __launch_bounds__(256) void k_bn_stats(const float* __restrict__ Y,
                                                  int M, int C, float* __restrict__ stats) {
  int c = blockIdx.x;
  __shared__ float s1[256], s2[256];
  float a = 0.f, b = 0.f;
  for (int m = threadIdx.x; m < M; m += 256) {
    float v = Y[(size_t)m * C + c];
    a += v; b += v * v;
  }
  s1[threadIdx.x] = a; s2[threadIdx.x] = b;
  __syncthreads();
  for (int off = 128; off > 0; off >>= 1) {
    if ((int)threadIdx.x < off) {
      s1[threadIdx.x] += s1[threadIdx.x + off];
      s2[threadIdx.x] += s2[threadIdx.x + off];
    }
    __syncthreads();
  }
  if (threadIdx.x == 0) {
    float mean = s1[0] / (float)M;
    stats[c] = mean;
    stats[512 + c] = s2[0] / (float)M - mean * mean;
  }
}

// ---------------- normalize * gamma + beta, ReLU ----------------
__global__ void k_bn_relu(float* __restrict__ Y, int tot, int C,
                          const float* __restrict__ stats,
                          const float* __restrict__ g, const float* __restrict__ be) {
  int i = blockIdx.x * blockDim.x + threadIdx.x;
  if (i >= tot) return;
  int c = i % C;
  float v = (Y[i] - stats[c]) * rsqrtf(stats[512 + c] + BN_EPS) * g[c] + be[c];
  Y[i] = fmaxf(v, 0.f);
}

// ---------------- max over the K (group) axis ----------------
__global__ void k_maxpool(const float* __restrict__ in, float* __restrict__ out,
                          int B, int S, int K, int C) {
  int i = blockIdx.x * blockDim.x + threadIdx.x;
  if (i >= B * S * C) return;
  int c = i % C; int t = i / C;
  int s = t % S; int b = t / S;
  float m = -3.4e38f;
  const float* base = in + (((size_t)b * S + s) * K) * C + c;
  for (int k = 0; k < K; ++k) m = fmaxf(m, base[(size_t)k * C]);
  out[i] = m;
}

// ---------------- broadcast global feature + concat skip (padded stride) ----------------
__global__ void k_bcast_concat(const float* __restrict__ f2, const float* __restrict__ f1,
                               float* __restrict__ out, int B, int S, int C2, int C1,
                               int Ctp) {
  int i = blockIdx.x * blockDim.x + threadIdx.x;
  if (i >= B * S * Ctp) return;
  int cc = i % Ctp; int t = i / Ctp;
  int s = t % S; int b = t / S;
  float v;
  if (cc < C2)            v = f2[(size_t)b * C2 + cc];
  else if (cc < C2 + C1)  v = f1[((size_t)b * S + s) * C1 + (cc - C2)];
  else                    v = 0.f;
  out[i] = v;
}

// ---------------- 3-NN search + inverse-distance weights ----------------
__global__ void k_knn3(const float* __restrict__ x1, const float* __restrict__ x2,
                       int* __restrict__ oi, float* __restrict__ ow,
                       int B, int S1, int S2) {
  int i = blockIdx.x * blockDim.x + threadIdx.x;
  if (i >= B * S1) return;
  int b = i / S1;
  float px = x1[(size_t)i * 3], py = x1[(size_t)i * 3 + 1], pz = x1[(size_t)i * 3 + 2];
  const float* q = x2 + (size_t)b * S2 * 3;
  float d0 = 3.4e38f, d1 = 3.4e38f, d2v = 3.4e38f;
  int i0 = 0, i1 = 0, i2 = 0;
  for (int n = 0; n < S2; ++n) {
    float dx = q[n * 3] - px, dy = q[n * 3 + 1] - py, dz = q[n * 3 + 2] - pz;
    float d = dx * dx + dy * dy + dz * dz;
    if (d < d0)       { d2v = d1; i2 = i1; d1 = d0; i1 = i0; d0 = d; i0 = n; }
    else if (d < d1)  { d2v = d1; i2 = i1; d1 = d;  i1 = n; }
    else if (d < d2v) { d2v = d;  i2 = n; }
  }
  float w0 = 1.f / (fmaxf(d0, 0.f) + 1e-8f);
  float w1 = 1.f / (fmaxf(d1, 0.f) + 1e-8f);
  float w2 = 1.f / (fmaxf(d2v, 0.f) + 1e-8f);
  float sw = w0 + w1 + w2;
  ow[(size_t)i * 3] = w0 / sw; ow[(size_t)i * 3 + 1] = w1 / sw; ow[(size_t)i * 3 + 2] = w2 / sw;
  oi[(size_t)i * 3] = i0; oi[(size_t)i * 3 + 1] = i1; oi[(size_t)i * 3 + 2] = i2;
}

// ---------------- weighted interpolation + concat skip (padded stride) ----------------
__global__ void k_interp_concat(const float* __restrict__ f2, const int* __restrict__ ki,
                                const float* __restrict__ kw, const float* __restrict__ f1,
                                float* __restrict__ out,
                                int B, int S1, int S2, int C2, int C1, int Ctp) {
  int i = blockIdx.x * blockDim.x + threadIdx.x;
  if (i >= B * S1 * Ctp) return;
  int cc = i % Ctp; int t = i / Ctp;
  int s = t % S1; int b = t / S1;
  float v;
  if (cc < C2) {
    size_t r = (size_t)b * S1 + s;
    const int* id = ki + r * 3;
    const float* w = kw + r * 3;
    const float* fb = f2 + (size_t)b * S2 * C2;
    v = w[0] * fb[(size_t)id[0] * C2 + cc] +
        w[1] * fb[(size_t)id[1] * C2 + cc] +
        w[2] * fb[(size_t)id[2] * C2 + cc];
  } else if (cc < C2 + C1) {
    v = f1[((size_t)b * S1 + s) * C1 + (cc - C2)];
  } else {
    v = 0.f;
  }
  out[i] = v;
}

// ---------------- row-wise log_softmax (input may have padded leading dim) ----------------
__global__ void k_log_softmax(const float* __restrict__ in, float* __restrict__ out,
                              int rows, int C, int ld) {
  int r = blockIdx.x * blockDim.x + threadIdx.x;
  if (r >= rows) return;
  const float* z = in + (size_t)r * ld;
  float mx = z[0];
  for (int c = 1; c < C; ++c) mx = fmaxf(mx, z[c]);
  float s = 0.f;
  for (int c = 0; c < C; ++c) s += expf(z[c] - mx);
  float ls = mx + logf(s);
  for (int c = 0; c < C; ++c) out[(size_t)r * C + c] = z[c] - ls;
}

__global__ void k_copy(const float* __restrict__ in, float* __restrict__ out, int n) {
  int i = blockIdx.x * blockDim.x + threadIdx.x;
  if (i < n) out[i] = in[i];
}

// ---------------- host-side orchestration ----------------
struct PLay {
  const float *wp, *bp, *g, *be;  // padded weight/bias, raw gamma/beta
  int Cin, Cout, Cinp, Coutp;
};

static inline void gemm(const float* X, int M, const PLay& L, float* Y, hipStream_t s) {
  dim3 grid(L.Coutp >> 4, CDIV(M >> 6, 8));
  size_t shmem = (size_t)(L.Cinp << 4) * sizeof(float);   // 16-row weight tile
  k_gemm_bias<<<grid, 256, shmem, s>>>(X, L.wp, L.bp, Y, M, L.Cinp, L.Coutp);
}

static inline void mlp_layer(const float* X, int M, const PLay& L, float* Y,
                             float* stats, hipStream_t s) {
  gemm(X, M, L, Y, s);
  k_bn_stats<<<L.Cout, 256, 0, s>>>(Y, M, L.Cout, stats);
  int tot = M * L.Cout;
  k_bn_relu<<<CDIV(tot, 256), 256, 0, s>>>(Y, tot, L.Cout, stats, L.g, L.be);
}

extern "C" void kernel_launch(void* const* d_in, const int* in_sizes, int n_in,
                              void* d_out, int out_size, void* d_ws, size_t ws_size,
                              hipStream_t stream) {
  (void)in_sizes; (void)n_in; (void)out_size; (void)ws_size;
  const int B = 16, N = 4096;
  const float* xyz = (const float*)d_in[0];

  float* W     = (float*)d_ws;
  float* pts   = W + OFF_PTS;
  float* dist  = W + OFF_DIST;
  int*   fidx  = (int*)(W + OFF_FIDX);
  float* l1x   = W + OFF_L1X;
  int*   gidx1 = (int*)(W + OFF_GIDX1);
  float* l1f   = W + OFF_L1F;
  float* l2x   = W + OFF_L2X;
  int*   gidx2 = (int*)(W + OFF_GIDX2);
  float* l2f   = W + OFF_L2F;
  float* l3f   = W + OFF_L3F;
  float* n2    = W + OFF_N2;
  int*   ki2   = (int*)(W + OFF_KI2);
  float* kw2   = W + OFF_KW2;
  float* n1    = W + OFF_N1;
  int*   ki3   = (int*)(W + OFF_KI3);
  float* kw3   = W + OFF_KW3;
  float* stats = W + OFF_STATS;
  float* wpad  = W + OFF_WPAD;
  float* BUF0  = W + OFF_BUF0;
  float* BUF1  = W + OFF_BUF1;

  // ---- build padded parameter set (17 linear layers; last one is head2 w/o BN)
  const int NL = 17;
  const int dims[NL][2] = {
    {3, 32}, {32, 32}, {32, 64},          // lem1
    {64, 64}, {64, 64}, {64, 128},        // lem2
    {128, 128}, {128, 256}, {256, 512},   // lem3
    {640, 256}, {256, 256},               // usm1
    {320, 256}, {256, 256},               // usm2
    {259, 128}, {128, 128},               // usm3
    {128, 128},                           // head1
    {128, 13}                             // head2
  };
  PLay lay[NL];
  size_t cur = 0;
  int p = 1;
  for (int l = 0; l < NL; ++l) {
    int Cin = dims[l][0], Cout = dims[l][1];
    int Cinp  = CDIV(Cin, 16) * 16;        // K chunked by 16
    int Coutp = CDIV(Cout, 16) * 16;
    const float* wr = (const float*)d_in[p];
    const float* br = (const float*)d_in[p + 1];
    if (l < NL - 1) {  // BN layers have 4 params
      lay[l].g  = (const float*)d_in[p + 2];
      lay[l].be = (const float*)d_in[p + 3];
      p += 4;
    } else {           // head2: w, b only
      lay[l].g = lay[l].be = nullptr;
      p += 2;
    }
    float* wp = wpad + cur; cur += (size_t)Coutp * Cinp;
    float* bp = wpad + cur; cur += (size_t)Coutp;
    k_pad_wb<<<CDIV(Coutp * Cinp, 256), 256, 0, stream>>>(wr, br, wp, bp, Cout, Cin, Coutp, Cinp);
    lay[l].wp = wp; lay[l].bp = bp;
    lay[l].Cin = Cin; lay[l].Cout = Cout; lay[l].Cinp = Cinp; lay[l].Coutp = Coutp;
  }

  k_transpose_xyz<<<CDIV(B * N * 3, 256), 256, 0, stream>>>(xyz, pts, B, N);

  // ---- SA1: 4096 -> 1024 centers, r=0.1, K=32, MLP 3(->16 padded)->32->32->64
  k_fps<<<B, 256, 0, stream>>>(pts, dist, fidx, N, 1024);
  k_gather_xyz<<<CDIV(B * 1024 * 3, 256), 256, 0, stream>>>(pts, fidx, l1x, B, N, 1024);
  k_ball_query<<<B, 256, 0, stream>>>(pts, l1x, gidx1, N, 1024, 0.01f, 32);
  k_gather_group<<<CDIV(B * 1024 * 32 * 16, 256), 256, 0, stream>>>(pts, gidx1, BUF0, B, N, 1024, 32, 3, 16);
  mlp_layer(BUF0, B * 1024 * 32, lay[0], BUF1, stats, stream);
  mlp_layer(BUF1, B * 1024 * 32, lay[1], BUF0, stats, stream);
  mlp_layer(BUF0, B * 1024 * 32, lay[2], BUF1, stats, stream);
  k_maxpool<<<CDIV(B * 1024 * 64, 256), 256, 0, stream>>>(BUF1, l1f, B, 1024, 32, 64);

  // ---- SA2: 1024 -> 256 centers, r=0.2, K=32, MLP 64->64->64->128
  k_fps<<<B, 256, 0, stream>>>(l1x, dist, fidx, 1024, 256);
  k_gather_xyz<<<CDIV(B * 256 * 3, 256), 256, 0, stream>>>(l1x, fidx, l2x, B, 1024, 256);
  k_ball_query<<<B, 256, 0, stream>>>(l1x, l2x, gidx2, 1024, 256, 0.04f, 32);
  k_gather_group<<<CDIV(B * 256 * 32 * 64, 256), 256, 0, stream>>>(l1f, gidx2, BUF0, B, 1024, 256, 32, 64, 64);
  mlp_layer(BUF0, B * 256 * 32, lay[3], BUF1, stats, stream);
  mlp_layer(BUF1, B * 256 * 32, lay[4], BUF0, stats, stream);
  mlp_layer(BUF0, B * 256 * 32, lay[5], BUF1, stats, stream);
  k_maxpool<<<CDIV(B * 256 * 128, 256), 256, 0, stream>>>(BUF1, l2f, B, 256, 32, 128);

  // ---- SA-all: MLP 128->128->256->512 over all 256 points, global max
  mlp_layer(l2f,  B * 256, lay[6], BUF0, stats, stream);
  mlp_layer(BUF0, B * 256, lay[7], BUF1, stats, stream);
  mlp_layer(BUF1, B * 256, lay[8], BUF0, stats, stream);
  k_maxpool<<<CDIV(B * 512, 256), 256, 0, stream>>>(BUF0, l3f, B, 1, 256, 512);

  // ---- FP1 (usm1): broadcast l3f (512) ++ l2f (128) -> MLP 640->256->256
  k_bcast_concat<<<CDIV(B * 256 * 640, 256), 256, 0, stream>>>(l3f, l2f, BUF0, B, 256, 512, 128, 640);
  mlp_layer(BUF0, B * 256, lay[9],  BUF1, stats, stream);
  mlp_layer(BUF1, B * 256, lay[10], n2,   stats, stream);

  // ---- FP2 (usm2): interp n2 to 1024 pts ++ l1f (64) -> MLP 320->256->256
  k_knn3<<<CDIV(B * 1024, 256), 256, 0, stream>>>(l1x, l2x, ki2, kw2, B, 1024, 256);
  k_interp_concat<<<CDIV(B * 1024 * 320, 256), 256, 0, stream>>>(n2, ki2, kw2, l1f, BUF0, B, 1024, 256, 256, 64, 320);
  mlp_layer(BUF0, B * 1024, lay[11], BUF1, stats, stream);
  mlp_layer(BUF1, B * 1024, lay[12], n1,   stats, stream);

  // ---- FP3 (usm3): interp n1 to 4096 pts ++ xyz (3) -> MLP 259(->272 padded)->128->128
  k_knn3<<<CDIV(B * 4096, 256), 256, 0, stream>>>(pts, l1x, ki3, kw3, B, 4096, 1024);
  k_interp_concat<<<CDIV(B * 4096 * 272, 256), 256, 0, stream>>>(n1, ki3, kw3, pts, BUF0, B, 4096, 1024, 256, 3, 272);
  mlp_layer(BUF0, B * 4096, lay[13], BUF1, stats, stream);
  mlp_layer(BUF1, B * 4096, lay[14], BUF0, stats, stream);

  // ---- head: 128 -> 128 (BN+ReLU), 128 -> 13 (linear, padded to 16), log_softmax
  mlp_layer(BUF0, B * 4096, lay[15], BUF1, stats, stream);
  gemm(BUF1, B * 4096, lay[16], BUF0, stream);   // Y stride = 16
  k_log_softmax<<<CDIV(B * 4096, 256), 256, 0, stream>>>(BUF0, (float*)d_out, B * 4096, 13, 16);
  // second output: l3f transposed (B,512,1) == flat copy of l3f
  k_copy<<<CDIV(B * 512, 256), 256, 0, stream>>>(l3f, (float*)d_out + (size_t)B * 4096 * 13, B * 512);
}